// Render_50036368998876
// MI455X (gfx1250) — compile-verified
//
#include <hip/hip_runtime.h>

#define FAR_DIST 1e10f
#define EPS      1e-10f
#define NSAMP    192
#define WAVE     32
#define RAYS_PER_BLOCK 8
#define NCHUNK   (NSAMP / WAVE)   // 6

// One wave32 per ray. Lanes hold 32 consecutive samples; 6 chunks per ray.
// rgb_raw (the 151MB stream) is staged global->LDS with the CDNA5 async copy
// engine (ASYNCcnt), double-buffered 2 chunks ahead of the serial scan.
__global__ __launch_bounds__(RAYS_PER_BLOCK * WAVE)
void nerf_render_kernel(const float* __restrict__ rgb_raw,
                        const float* __restrict__ density,
                        const float* __restrict__ z_vals,
                        const float* __restrict__ dir,
                        float* __restrict__ out_rgb,
                        float* __restrict__ out_w,
                        float* __restrict__ out_depth,
                        float* __restrict__ out_acc,
                        float* __restrict__ out_disp,
                        int B)
{
    // 16B slots so the LDS read back is an aligned ds_load_b128 (4th lane-float unused)
    __shared__ float4 smem[2][RAYS_PER_BLOCK][WAVE];

    const int wid  = (int)(threadIdx.x >> 5);
    const int lane = (int)(threadIdx.x & 31);
    const int b    = (int)blockIdx.x * RAYS_PER_BLOCK + wid;
    if (b >= B) return;

    const int base = b * NSAMP;

    // ray-direction norm (wave-uniform)
    const float dx = dir[b * 3 + 0], dy = dir[b * 3 + 1], dz = dir[b * 3 + 2];
    const float norm = sqrtf(dx * dx + dy * dy + dz * dz);

    // Issue 3 async b32 copies (12B = one rgb sample per lane) into this wave's slot.
    // Leading s_wait_dscnt 0 guarantees any pending ds_load of the recycled buffer
    // has completed before the async engine may overwrite it.
    auto issue_rgb = [&](int c) {
        unsigned goff  = (unsigned)((base + c * WAVE + lane) * 3) * 4u;
        unsigned laddr = (unsigned)(size_t)&smem[c & 1][wid][lane]; // low 32b of flat LDS addr = LDS offset
        asm volatile(
            "s_wait_dscnt 0x0\n\t"
            "global_load_async_to_lds_b32 %0, %1, %2 offset:0\n\t"
            "global_load_async_to_lds_b32 %0, %1, %2 offset:4\n\t"
            "global_load_async_to_lds_b32 %0, %1, %2 offset:8"
            :: "v"(laddr), "v"(goff), "s"(rgb_raw) : "memory");
    };

    issue_rgb(0);
    issue_rgb(1);

    float carry = 1.0f;                      // product of (1-alpha+eps) over prior chunks
    float cr = 0.f, cg = 0.f, cb = 0.f;      // rgb accumulators
    float cz = 0.f, ca = 0.f;                // depth / acc accumulators

    for (int c = 0; c < NCHUNK; ++c) {
        const int n   = c * WAVE + lane;
        const int idx = base + n;

        // one-shot streams: non-temporal so we don't churn L2
        const float z = __builtin_nontemporal_load(&z_vals[idx]);
        const float d = __builtin_nontemporal_load(&density[idx]);

        // z[n+1]: neighbor via shuffle; lane 31 fetches the next chunk's first z
        float znext = __shfl_down(z, 1, WAVE);
        float z31 = 0.0f;
        if (lane == 31 && c < NCHUNK - 1) z31 = z_vals[idx + 1];
        const float zp1  = (lane == 31) ? z31 : znext;
        const float dist = (n == NSAMP - 1) ? FAR_DIST * norm : (zp1 - z) * norm;

        const float a     = fmaxf(d, 0.0f);
        const float e     = __expf(-a * dist);
        const float alpha = 1.0f - e;
        const float f     = e + EPS;         // 1 - alpha + eps

        // inclusive prefix product across the wave (5-step multiplicative scan)
        float p = f;
        #pragma unroll
        for (int off = 1; off < WAVE; off <<= 1) {
            float t = __shfl_up(p, off, WAVE);
            if (lane >= off) p *= t;
        }
        float ex = __shfl_up(p, 1, WAVE);    // exclusive scan
        if (lane == 0) ex = 1.0f;
        const float trans = carry * ex;
        const float w     = alpha * trans;
        carry *= __shfl(p, WAVE - 1, WAVE);  // chunk total product -> carry

        // wait for this chunk's staged rgb (next chunk's 3 copies may stay in flight)
        if (c < NCHUNK - 1) asm volatile("s_wait_asynccnt 0x3" ::: "memory");
        else                asm volatile("s_wait_asynccnt 0x0" ::: "memory");
        const float4 rv = smem[c & 1][wid][lane];

        if (c + 2 < NCHUNK) issue_rgb(c + 2);

        const float sr = 1.0f / (1.0f + __expf(-rv.x));
        const float sg = 1.0f / (1.0f + __expf(-rv.y));
        const float sb = 1.0f / (1.0f + __expf(-rv.z));

        cr += w * sr;  cg += w * sg;  cb += w * sb;
        cz += w * z;   ca += w;

        __builtin_nontemporal_store(w, &out_w[idx]);
    }

    // wave reductions for the five per-ray outputs
    #pragma unroll
    for (int off = 16; off >= 1; off >>= 1) {
        cr += __shfl_down(cr, off, WAVE);
        cg += __shfl_down(cg, off, WAVE);
        cb += __shfl_down(cb, off, WAVE);
        cz += __shfl_down(cz, off, WAVE);
        ca += __shfl_down(ca, off, WAVE);
    }
    if (lane == 0) {
        __builtin_nontemporal_store(cr, &out_rgb[b * 3 + 0]);
        __builtin_nontemporal_store(cg, &out_rgb[b * 3 + 1]);
        __builtin_nontemporal_store(cb, &out_rgb[b * 3 + 2]);
        out_depth[b] = cz;
        out_acc[b]   = ca;
        out_disp[b]  = 1.0f / fmaxf(EPS, cz / ca);
    }
}

extern "C" void kernel_launch(void* const* d_in, const int* in_sizes, int n_in,
                              void* d_out, int out_size, void* d_ws, size_t ws_size,
                              hipStream_t stream) {
    (void)n_in; (void)out_size; (void)d_ws; (void)ws_size;
    const float* rgb_raw = (const float*)d_in[0];   // [B, N, 3]
    const float* density = (const float*)d_in[1];   // [B, N, 1]
    const float* z_vals  = (const float*)d_in[2];   // [B, N]
    const float* dir     = (const float*)d_in[3];   // [B, 3]

    const int B = in_sizes[2] / NSAMP;

    float* out       = (float*)d_out;
    float* out_rgb   = out;                                   // [B,3]
    float* out_w     = out_rgb + (size_t)B * 3;               // [B,N]
    float* out_depth = out_w + (size_t)B * NSAMP;             // [B]
    float* out_acc   = out_depth + B;                         // [B]
    float* out_disp  = out_acc + B;                           // [B]

    const dim3 block(RAYS_PER_BLOCK * WAVE);
    const dim3 grid((B + RAYS_PER_BLOCK - 1) / RAYS_PER_BLOCK);
    nerf_render_kernel<<<grid, block, 0, stream>>>(
        rgb_raw, density, z_vals, dir,
        out_rgb, out_w, out_depth, out_acc, out_disp, B);
}